// DetectionLoss_41180146434282
// MI455X (gfx1250) — compile-verified
//
#include <hip/hip_runtime.h>

// Detection loss (focal + hard-negative mining via radix-select + GIoU/smooth-L1)
// for MI455X / gfx1250. B=16 images, A=65536 anchors, T=32 targets, C=21 classes.
// Inputs (setup_inputs order): conf_pred f32[B,A,C], bbox_pred f32[B,A,4],
// anchors f32[A,4], target_boxes f32[B,T,4], target_labels i32[B,T].
// Output f32[3] = {total, conf_loss, bbox_loss}.

#define B_IMG 16
#define A_N   65536
#define T_N   32
#define C_N   21
#define POS_THR 0.5f
#define NEG_THR 0.4f
#define EPS_F   1e-6f

typedef float v8f __attribute__((ext_vector_type(8)));
typedef float v2f __attribute__((ext_vector_type(2)));

struct Ctl {
  unsigned long long bestA[B_IMG * T_N];   // packed (iou_bits<<32)|~anchor
  int      numPos[B_IMG];
  int      nNeg[B_IMG];
  float    posSum[B_IMG];
  float    totalSum[B_IMG];
  float    bboxSum[B_IMG];
  float    sumGt[B_IMG][2];
  unsigned prefix[B_IMG][2];
  int      remaining[B_IMG][2];
  unsigned cg[B_IMG][2];
  unsigned thrBits[B_IMG][2];
  int      kval[B_IMG][2];
  unsigned hist[B_IMG][2][2048];
};

// ---- wave reduction via matrix pipe: D = ones(16x4) x B(4x16) + C ----------
// Each of the 32 lane values lands in exactly one B slot, so every D row is the
// 16 column sums; summing lanes 0..15 of c[0] (4 shfl_xor steps) = total.
__device__ __forceinline__ float wave_sum_f32(float v) {
  v2f a = {1.0f, 1.0f};
  v2f bm = {v, 0.0f};
  v8f c = {0.f, 0.f, 0.f, 0.f, 0.f, 0.f, 0.f, 0.f};
  c = __builtin_amdgcn_wmma_f32_16x16x4_f32(false, a, false, bm, (short)0, c,
                                            false, false);
  float x = c[0];
  x += __shfl_xor(x, 1, 32);
  x += __shfl_xor(x, 2, 32);
  x += __shfl_xor(x, 4, 32);
  x += __shfl_xor(x, 8, 32);
  return x;  // all lanes hold the 32-lane total
}

// ---- shared focal math (identical bits in every pass) ----------------------
__device__ __forceinline__ void softmax_row(const float* __restrict__ row,
                                            float* e, float& inv) {
  float mx = row[0];
#pragma unroll
  for (int c = 1; c < C_N; ++c) mx = fmaxf(mx, row[c]);
  float den = 0.0f;
#pragma unroll
  for (int c = 0; c < C_N; ++c) { e[c] = expf(row[c] - mx); den += e[c]; }
  inv = 1.0f / den;
}

__device__ __forceinline__ float focal_cell(float p, bool is_pos_cell) {
  float pt  = is_pos_cell ? p : 1.0f - p;
  float af  = is_pos_cell ? 0.25f : 0.75f;
  float omp = 1.0f - pt;
  float fo  = -af * omp * omp * logf(fmaxf(pt, EPS_F));
  return fmaxf(fo, 0.0f);  // also maps -0 -> +0 (radix monotonicity)
}

// ---- K1: anchor<->target matching ------------------------------------------
__global__ void k_match(const float* __restrict__ anchors,
                        const float* __restrict__ tboxes,
                        Ctl* ctl, unsigned char* __restrict__ bestT,
                        unsigned char* __restrict__ flags) {
  __shared__ float4 stb[T_N];
  __shared__ unsigned long long sBest[T_N];
  const int b = blockIdx.y;
  const int a = blockIdx.x * blockDim.x + threadIdx.x;
  if (threadIdx.x < T_N) {
    stb[threadIdx.x] = ((const float4*)tboxes)[b * T_N + threadIdx.x];
    sBest[threadIdx.x] = 0ull;
  }
  __syncthreads();
  const float4 an = ((const float4*)anchors)[a];
  const float aArea = (an.z - an.x) * (an.w - an.y);
  float best = -1.0f;
  int bt = 0;
#pragma unroll 4
  for (int t = 0; t < T_N; ++t) {
    const float4 tb = stb[t];
    float x1 = fmaxf(an.x, tb.x), y1 = fmaxf(an.y, tb.y);
    float x2 = fminf(an.z, tb.z), y2 = fminf(an.w, tb.w);
    float inter = fmaxf(x2 - x1, 0.f) * fmaxf(y2 - y1, 0.f);
    float tArea = (tb.z - tb.x) * (tb.w - tb.y);
    float iou = inter / (aArea + tArea - inter + EPS_F);
    if (iou > best) { best = iou; bt = t; }  // first-index tie-break
    unsigned long long packed =
        ((unsigned long long)__float_as_uint(iou) << 32) |
        (unsigned)(~(unsigned)a);            // smaller index wins ties
    atomicMax(&sBest[t], packed);
  }
  const size_t ba = (size_t)b * A_N + a;
  bestT[ba] = (unsigned char)bt;
  unsigned char f = 0;
  if (best >= POS_THR) f |= 1;
  if (best <  NEG_THR) f |= 2;
  flags[ba] = f;
  __syncthreads();
  if (threadIdx.x < T_N)
    atomicMax(&ctl->bestA[b * T_N + threadIdx.x], sBest[threadIdx.x]);
}

// ---- K2: force-match best anchor per target --------------------------------
__global__ void k_force(Ctl* ctl, unsigned char* __restrict__ flags) {
  int i = threadIdx.x;  // B*T = 512
  int b = i / T_N;
  unsigned a = ~(unsigned)(ctl->bestA[i] & 0xFFFFFFFFull);
  flags[(size_t)b * A_N + a] = 1;  // pos=1, neg=0 (idempotent)
}

// ---- K3: counts + bbox loss ------------------------------------------------
__global__ void k_count_bbox(const float* __restrict__ bbox_pred,
                             const float* __restrict__ tboxes,
                             const unsigned char* __restrict__ flags,
                             const unsigned char* __restrict__ bestT,
                             Ctl* ctl) {
  __shared__ float4 stb[T_N];
  const int b = blockIdx.y;
  const int a = blockIdx.x * blockDim.x + threadIdx.x;
  if (threadIdx.x < T_N)
    stb[threadIdx.x] = ((const float4*)tboxes)[b * T_N + threadIdx.x];
  __syncthreads();
  const size_t ba = (size_t)b * A_N + a;
  const unsigned char f = flags[ba];
  const int isPos = f & 1, isNeg = (f >> 1) & 1;
  float bl = 0.0f;
  if (isPos) {
    const float4 p = ((const float4*)bbox_pred)[ba];
    const float4 m = stb[bestT[ba]];
    float x1 = fmaxf(p.x, m.x), y1 = fmaxf(p.y, m.y);
    float x2 = fminf(p.z, m.z), y2 = fminf(p.w, m.w);
    float inter = fmaxf(x2 - x1, 0.f) * fmaxf(y2 - y1, 0.f);
    float a1 = (p.z - p.x) * (p.w - p.y);
    float a2 = (m.z - m.x) * (m.w - m.y);
    float uni = a1 + a2 - inter;
    float iou = inter / (uni + EPS_F);
    float ex1 = fminf(p.x, m.x), ey1 = fminf(p.y, m.y);
    float ex2 = fmaxf(p.z, m.z), ey2 = fmaxf(p.w, m.w);
    float enc = (ex2 - ex1) * (ey2 - ey1);
    float giou = iou - (enc - uni) / (enc + EPS_F);
    float d0 = p.x - m.x, d1 = p.y - m.y, d2 = p.z - m.z, d3 = p.w - m.w;
    float l1 = 0.0f;
    float ad;
    ad = fabsf(d0); l1 += (ad < 1.0f) ? 0.5f * d0 * d0 : ad - 0.5f;
    ad = fabsf(d1); l1 += (ad < 1.0f) ? 0.5f * d1 * d1 : ad - 0.5f;
    ad = fabsf(d2); l1 += (ad < 1.0f) ? 0.5f * d2 * d2 : ad - 0.5f;
    ad = fabsf(d3); l1 += (ad < 1.0f) ? 0.5f * d3 * d3 : ad - 0.5f;
    l1 *= 0.25f;
    bl = (1.0f - giou) + 0.5f * l1;
  }
  unsigned long long bp = __ballot(isPos);
  unsigned long long bn = __ballot(isNeg);
  float ws = wave_sum_f32(bl);
  if ((threadIdx.x & 31) == 0) {
    atomicAdd(&ctl->numPos[b], (int)__popcll(bp));
    atomicAdd(&ctl->nNeg[b],  (int)__popcll(bn));
    atomicAdd(&ctl->bboxSum[b], ws);
  }
}

// ---- K4: focal pass 1: sums + level-1 histogram (bits[31:21]) --------------
__global__ void k_focal1(const float* __restrict__ conf,
                         const int* __restrict__ tlabels,
                         const unsigned char* __restrict__ flags,
                         const unsigned char* __restrict__ bestT,
                         Ctl* ctl) {
  __shared__ unsigned shist[2048];
  __shared__ int stl[T_N];
  const int b = blockIdx.y;
  for (int i = threadIdx.x; i < 2048; i += blockDim.x) shist[i] = 0;
  if (threadIdx.x < T_N) stl[threadIdx.x] = tlabels[b * T_N + threadIdx.x];
  __syncthreads();
  const int a = blockIdx.x * blockDim.x + threadIdx.x;
  const size_t ba = (size_t)b * A_N + a;
  const unsigned char f = flags[ba];
  const bool isPos = (f & 1) != 0, isNeg = (f & 2) != 0;
  const int posCls = stl[bestT[ba]];
  const float* row = conf + ba * C_N;
  float e[C_N], inv;
  softmax_row(row, e, inv);
  float tSum = 0.0f, pSum = 0.0f;
#pragma unroll
  for (int c = 0; c < C_N; ++c) {
    float fo = focal_cell(e[c] * inv, isPos && (c == posCls));
    tSum += fo;
    if (isPos) pSum += fo;
    if (isNeg) atomicAdd(&shist[__float_as_uint(fo) >> 21], 1u);
  }
  float wt = wave_sum_f32(tSum);
  float wp = wave_sum_f32(pSum);
  if ((threadIdx.x & 31) == 0) {
    atomicAdd(&ctl->totalSum[b], wt);
    atomicAdd(&ctl->posSum[b], wp);
  }
  __syncthreads();
  for (int i = threadIdx.x; i < 2048; i += blockDim.x)
    if (shist[i]) atomicAdd(&ctl->hist[b][0][i], shist[i]);
}

// ---- K5: scan level 1 (both selections share one histogram) ----------------
__global__ void k_scan1(Ctl* ctl) {
  int b = threadIdx.x;
  if (b >= B_IMG) return;
  int np = ctl->numPos[b], nn = ctl->nNeg[b];
  int kk[2];
  kk[0] = min(3 * np, nn);
  kk[1] = min(100, nn);
  for (int s = 0; s < 2; ++s) {
    ctl->kval[b][s] = kk[s];
    int r = kk[s];
    if (r <= 0) {
      ctl->prefix[b][s] = 0xFFFFFFFFu;  // matches no value in later passes
      ctl->thrBits[b][s] = 0xFFFFFFFFu;
      ctl->remaining[b][s] = 0;
      ctl->cg[b][s] = 0;
      continue;
    }
    unsigned cgl = 0;
    int bucket = 0;
    for (int i = 2047; i >= 0; --i) {
      unsigned c = ctl->hist[b][0][i];
      if (cgl + c >= (unsigned)r) { bucket = i; break; }
      cgl += c;
    }
    ctl->prefix[b][s] = (unsigned)bucket << 21;
    ctl->remaining[b][s] = r - (int)cgl;
    ctl->cg[b][s] = cgl;
  }
  for (int i = 0; i < 2048; ++i) { ctl->hist[b][0][i] = 0; ctl->hist[b][1][i] = 0; }
}

// ---- K6: focal pass 2: level-2 histograms (bits[20:10]), neg anchors only --
__global__ void k_focal2(const float* __restrict__ conf,
                         const unsigned char* __restrict__ flags, Ctl* ctl) {
  __shared__ unsigned h[2][2048];
  const int b = blockIdx.y;
  for (int i = threadIdx.x; i < 4096; i += blockDim.x) (&h[0][0])[i] = 0;
  __syncthreads();
  const int a = blockIdx.x * blockDim.x + threadIdx.x;
  const size_t ba = (size_t)b * A_N + a;
  const unsigned pre0 = ctl->prefix[b][0], pre1 = ctl->prefix[b][1];
  if (flags[ba] & 2) {  // neg anchors: is_pos_cell always false
    const float* row = conf + ba * C_N;
    float e[C_N], inv;
    softmax_row(row, e, inv);
#pragma unroll
    for (int c = 0; c < C_N; ++c) {
      unsigned bits = __float_as_uint(focal_cell(e[c] * inv, false));
      unsigned hi = bits & 0xFFE00000u, mid = (bits >> 10) & 0x7FFu;
      if (hi == pre0) atomicAdd(&h[0][mid], 1u);
      if (hi == pre1) atomicAdd(&h[1][mid], 1u);
    }
  }
  __syncthreads();
  for (int i = threadIdx.x; i < 2048; i += blockDim.x) {
    if (h[0][i]) atomicAdd(&ctl->hist[b][0][i], h[0][i]);
    if (h[1][i]) atomicAdd(&ctl->hist[b][1][i], h[1][i]);
  }
}

// ---- K7: scan level 2 ------------------------------------------------------
__global__ void k_scan2(Ctl* ctl) {
  int b = threadIdx.x;
  if (b >= B_IMG) return;
  for (int s = 0; s < 2; ++s) {
    if (ctl->kval[b][s] <= 0) continue;
    int r = ctl->remaining[b][s];
    unsigned cgl = 0;
    int bucket = 0;
    for (int i = 2047; i >= 0; --i) {
      unsigned c = ctl->hist[b][s][i];
      if (cgl + c >= (unsigned)r) { bucket = i; break; }
      cgl += c;
    }
    ctl->prefix[b][s] |= (unsigned)bucket << 10;
    ctl->remaining[b][s] = r - (int)cgl;
    ctl->cg[b][s] += cgl;
  }
  for (int i = 0; i < 2048; ++i) { ctl->hist[b][0][i] = 0; ctl->hist[b][1][i] = 0; }
}

// ---- K8: focal pass 3: level-3 histograms (bits[9:0]) ----------------------
__global__ void k_focal3(const float* __restrict__ conf,
                         const unsigned char* __restrict__ flags, Ctl* ctl) {
  __shared__ unsigned h[2][1024];
  const int b = blockIdx.y;
  for (int i = threadIdx.x; i < 2048; i += blockDim.x) (&h[0][0])[i] = 0;
  __syncthreads();
  const int a = blockIdx.x * blockDim.x + threadIdx.x;
  const size_t ba = (size_t)b * A_N + a;
  const unsigned pre0 = ctl->prefix[b][0], pre1 = ctl->prefix[b][1];
  if (flags[ba] & 2) {
    const float* row = conf + ba * C_N;
    float e[C_N], inv;
    softmax_row(row, e, inv);
#pragma unroll
    for (int c = 0; c < C_N; ++c) {
      unsigned bits = __float_as_uint(focal_cell(e[c] * inv, false));
      unsigned hi = bits & 0xFFFFFC00u, lo = bits & 0x3FFu;
      if (hi == pre0) atomicAdd(&h[0][lo], 1u);
      if (hi == pre1) atomicAdd(&h[1][lo], 1u);
    }
  }
  __syncthreads();
  for (int i = threadIdx.x; i < 1024; i += blockDim.x) {
    if (h[0][i]) atomicAdd(&ctl->hist[b][0][i], h[0][i]);
    if (h[1][i]) atomicAdd(&ctl->hist[b][1][i], h[1][i]);
  }
}

// ---- K9: scan level 3 -> exact thresholds ----------------------------------
__global__ void k_scan3(Ctl* ctl) {
  int b = threadIdx.x;
  if (b >= B_IMG) return;
  for (int s = 0; s < 2; ++s) {
    if (ctl->kval[b][s] <= 0) continue;
    int r = ctl->remaining[b][s];
    unsigned cgl = 0;
    int bucket = 0;
    for (int i = 1023; i >= 0; --i) {
      unsigned c = ctl->hist[b][s][i];
      if (cgl + c >= (unsigned)r) { bucket = i; break; }
      cgl += c;
    }
    ctl->thrBits[b][s] = ctl->prefix[b][s] | (unsigned)bucket;
    ctl->cg[b][s] += cgl;  // total count strictly greater than threshold
    ctl->remaining[b][s] = r - (int)cgl;
  }
}

// ---- K10: sum of values strictly above each threshold ----------------------
__global__ void k_sum(const float* __restrict__ conf,
                      const unsigned char* __restrict__ flags, Ctl* ctl) {
  const int b = blockIdx.y;
  const int a = blockIdx.x * blockDim.x + threadIdx.x;
  const size_t ba = (size_t)b * A_N + a;
  const unsigned thr0 = ctl->thrBits[b][0], thr1 = ctl->thrBits[b][1];
  float acc0 = 0.0f, acc1 = 0.0f;
  if (flags[ba] & 2) {
    const float* row = conf + ba * C_N;
    float e[C_N], inv;
    softmax_row(row, e, inv);
#pragma unroll
    for (int c = 0; c < C_N; ++c) {
      float fo = focal_cell(e[c] * inv, false);
      unsigned bits = __float_as_uint(fo);
      if (bits > thr0) acc0 += fo;
      if (bits > thr1) acc1 += fo;
    }
  }
  float w0 = wave_sum_f32(acc0);
  float w1 = wave_sum_f32(acc1);
  if ((threadIdx.x & 31) == 0) {
    atomicAdd(&ctl->sumGt[b][0], w0);
    atomicAdd(&ctl->sumGt[b][1], w1);
  }
}

// ---- K11: finalize ---------------------------------------------------------
__global__ void k_final(Ctl* ctl, float* __restrict__ out) {
  __shared__ float sc[B_IMG], sb[B_IMG];
  const int b = threadIdx.x;
  if (b < B_IMG) {
    int np = ctl->numPos[b], nn = ctl->nNeg[b];
    int k0 = ctl->kval[b][0], k1 = ctl->kval[b][1];
    float top0 = 0.0f, top1 = 0.0f;
    if (k0 > 0)
      top0 = ctl->sumGt[b][0] +
             (float)(k0 - (int)ctl->cg[b][0]) * __uint_as_float(ctl->thrBits[b][0]);
    if (k1 > 0)
      top1 = ctl->sumGt[b][1] +
             (float)(k1 - (int)ctl->cg[b][1]) * __uint_as_float(ctl->thrBits[b][1]);
    float confL;
    if (np > 0) {
      if (nn > 0) {
        int den = np + k0; if (den < 1) den = 1;
        confL = (ctl->posSum[b] + top0) / (float)den;
      } else {
        int den = np < 1 ? 1 : np;
        confL = ctl->posSum[b] / (float)den;
      }
    } else {
      if (nn > 0) {
        int den = k1 < 1 ? 1 : k1;
        confL = top1 / (float)den;
      } else {
        confL = ctl->totalSum[b] / (float)(A_N * C_N);
      }
    }
    float bboxL = (np > 0) ? ctl->bboxSum[b] / (float)np : 0.0f;
    sc[b] = confL;
    sb[b] = bboxL;
  }
  __syncthreads();
  if (b == 0) {
    float c = 0.0f, bb = 0.0f;
    for (int i = 0; i < B_IMG; ++i) { c += sc[i]; bb += sb[i]; }
    c *= (1.0f / B_IMG);
    bb *= (1.0f / B_IMG);
    out[0] = c + bb;
    out[1] = c;
    out[2] = bb;
  }
}

extern "C" void kernel_launch(void* const* d_in, const int* in_sizes, int n_in,
                              void* d_out, int out_size, void* d_ws,
                              size_t ws_size, hipStream_t stream) {
  const float* conf    = (const float*)d_in[0];
  const float* bbox    = (const float*)d_in[1];
  const float* anchors = (const float*)d_in[2];
  const float* tbox    = (const float*)d_in[3];
  const int*   tlab    = (const int*)d_in[4];
  float* out = (float*)d_out;

  Ctl* ctl = (Ctl*)d_ws;
  size_t ctlBytes = (sizeof(Ctl) + 255) & ~(size_t)255;
  unsigned char* bestT = (unsigned char*)d_ws + ctlBytes;
  unsigned char* flags = bestT + (size_t)B_IMG * A_N;

  hipMemsetAsync(d_ws, 0, sizeof(Ctl), stream);

  dim3 grid(A_N / 256, B_IMG), blk(256);
  k_match     <<<grid, blk, 0, stream>>>(anchors, tbox, ctl, bestT, flags);
  k_force     <<<1, B_IMG * T_N, 0, stream>>>(ctl, flags);
  k_count_bbox<<<grid, blk, 0, stream>>>(bbox, tbox, flags, bestT, ctl);
  k_focal1    <<<grid, blk, 0, stream>>>(conf, tlab, flags, bestT, ctl);
  k_scan1     <<<1, 32, 0, stream>>>(ctl);
  k_focal2    <<<grid, blk, 0, stream>>>(conf, flags, ctl);
  k_scan2     <<<1, 32, 0, stream>>>(ctl);
  k_focal3    <<<grid, blk, 0, stream>>>(conf, flags, ctl);
  k_scan3     <<<1, 32, 0, stream>>>(ctl);
  k_sum       <<<grid, blk, 0, stream>>>(conf, flags, ctl);
  k_final     <<<1, 32, 0, stream>>>(ctl, out);
}